// knnLoss_39316130628224
// MI455X (gfx1250) — compile-verified
//
#include <hip/hip_runtime.h>
#include <hip/hip_bf16.h>
#include <math.h>
#include <stddef.h>

// ---------------------------------------------------------------------------
// KNN loss (K=5) between two 16384-point clouds on the WMMA pipe.
//
// d2(m,n) = |t_m|^2 + |s_n|^2 - 2 t_m.s_n is computed *entirely* inside
// V_WMMA_F32_16X16X4_F32 by packing the spare K=4 slot:
//     A row m  = ( tx, ty, tz, t2eff[m] )      (targets, 16x4)
//     B col n  = (-2sx,-2sy,-2sz, 1.0 )        (sources, 4x16)
//     C[m][n]  = t2eff[m] - 2 t.s   ->   d2 = s2 + C   (4 pk_add per tile)
// Invalid targets are masked by t2eff = 1e20 (sqrt -> 1e10, like reference).
// Top-5 selection runs on d^2 with a branchless 9-op insert, gated by a
// wave-uniform threshold test (v_min3 tree -> v_cmp -> s_cbranch), executed
// on the ~40 tiles/lane where the top-5 actually changes. EXEC stays all-1s
// around the WMMAs. All lane-half-dependent addressing is hoisted into
// per-lane strided pointers (4 x v_add_nc_u64 per 32 targets).
// ---------------------------------------------------------------------------

typedef float v2f __attribute__((ext_vector_type(2)));
typedef float v8f __attribute__((ext_vector_type(8)));

#define NPTS      16384           // 2 * 8192 flattened
#define KNN_K     5
#define TILES     (NPTS / 16)     // 1024 target tiles of 16
#define WAVES_PB  4               // 128 threads / block (4 wave32)
#define SRC_PB    (WAVES_PB * 16)
#define NBLOCKS   (NPTS / SRC_PB) // 256 blocks

// Branchless insert of x into ascending sorted 5-smallest {t0..t4}.
__device__ __forceinline__ void ins5(float& t0, float& t1, float& t2,
                                     float& t3, float& t4, float x) {
  float u;
  u = fmaxf(t0, x); t0 = fminf(t0, x);
  x = fmaxf(t1, u); t1 = fminf(t1, u);
  u = fmaxf(t2, x); t2 = fminf(t2, x);
  x = fmaxf(t3, u); t3 = fminf(t3, u);
  t4 = fminf(t4, x);
}

// |t|^2 per target; invalid (all-zero) targets baked to 1e20.
__global__ void knn_prep_t2(const float* __restrict__ tgt,
                            float* __restrict__ t2eff) {
  int m = blockIdx.x * blockDim.x + threadIdx.x;
  if (m >= NPTS) return;
  float x = tgt[3 * m + 0], y = tgt[3 * m + 1], z = tgt[3 * m + 2];
  bool valid = (x != 0.0f) || (y != 0.0f) || (z != 0.0f);
  t2eff[m] = valid ? (x * x + y * y + z * z) : 1e20f;
}

__global__ __launch_bounds__(WAVES_PB * 32)
void knn_main(const float* __restrict__ src, const float* __restrict__ tgt,
              const float* __restrict__ t2eff, float* __restrict__ partials) {
  const int lane = threadIdx.x & 31;
  const int wv   = threadIdx.x >> 5;
  const int lh   = lane >> 4;     // 0 => C rows 0..7 / K={0,1}; 1 => rows 8..15 / K={2,3}
  const int l16  = lane & 15;     // source column owned by this lane
  const int sIdx = blockIdx.x * SRC_PB + wv * 16 + l16;

  // Source point for this lane (column N of the WMMA output).
  const float sx = src[3 * sIdx + 0];
  const float sy = src[3 * sIdx + 1];
  const float sz = src[3 * sIdx + 2];
  const float s2 = sx * sx + sy * sy + sz * sz;
  const float wvalid =
      ((sx != 0.0f) || (sy != 0.0f) || (sz != 0.0f)) ? 1.0f : 0.0f;

  // B matrix 4x16 (KxN) f32: VGPR0 = K0 (lanes 0-15) / K2 (lanes 16-31),
  //                          VGPR1 = K1 / K3.  K3 carries the +1.0 for t2.
  v2f b;
  b.x = lh ? (-2.0f * sz) : (-2.0f * sx);
  b.y = lh ? 1.0f         : (-2.0f * sy);

  // Hoisted per-lane strided pointers for the A operands.
  //   x-stream : tx (half 0) or tz (half 1), stride 48 floats / 16 targets
  //   y-stream : ty (half 0) or t2eff (half 1)
  const float* px0 = tgt + (3 * l16 + 2 * lh);                   // tile 0
  const float* px1 = px0 + 48;                                   // tile 1
  const float* py0 = lh ? (t2eff + l16) : (tgt + 3 * l16 + 1);   // tile 0
  const float* py1 = lh ? (py0 + 16) : (py0 + 48);               // tile 1
  const ptrdiff_t stepx = 96;                 // +32 targets per iteration
  const ptrdiff_t stepy = lh ? 32 : 96;

  // Ascending top-5 of squared distance.
  float t0, t1, t2, t3, t4;
  t0 = t1 = t2 = t3 = t4 = 3.0e37f;

  for (int tt = 0; tt < TILES; tt += 2) {
    // Prefetch next iteration's coordinate stream (global_prefetch_b8).
    __builtin_prefetch(px0 + stepx, 0, 0);

    v2f a0, a1;
    a0.x = *px0; a0.y = *py0;
    a1.x = *px1; a1.y = *py1;
    px0 += stepx; px1 += stepx;
    py0 += stepy; py1 += stepy;

    v8f c0 = {}, c1 = {};
    c0 = __builtin_amdgcn_wmma_f32_16x16x4_f32(false, a0, false, b, (short)0,
                                               c0, false, false);
    c1 = __builtin_amdgcn_wmma_f32_16x16x4_f32(false, a1, false, b, (short)0,
                                               c1, false, false);

    // d2 = s2 + (t2 - 2*dot); rows r + 8*lh of each tile.
    float d[16];
#pragma unroll
    for (int r = 0; r < 8; ++r) {
      d[r]     = s2 + c0[r];
      d[8 + r] = s2 + c1[r];
    }

    // Wave-uniform gate: skip insertion unless some lane has a new top-5 entry.
    float mn = d[0];
#pragma unroll
    for (int r = 1; r < 16; ++r) mn = fminf(mn, d[r]);

    if (__ballot(mn < t4) != 0ull) {
#pragma unroll
      for (int r = 0; r < 16; ++r) ins5(t0, t1, t2, t3, t4, d[r]);
    }
  }

  // Merge the two lane-halves (same source column, disjoint target rows).
  // Snapshot partner values BEFORE inserting (lockstep hazard otherwise).
  const float o0 = __shfl_xor(t0, 16, 32);
  const float o1 = __shfl_xor(t1, 16, 32);
  const float o2 = __shfl_xor(t2, 16, 32);
  const float o3 = __shfl_xor(t3, 16, 32);
  const float o4 = __shfl_xor(t4, 16, 32);
  ins5(t0, t1, t2, t3, t4, o0);
  ins5(t0, t1, t2, t3, t4, o1);
  ins5(t0, t1, t2, t3, t4, o2);
  ins5(t0, t1, t2, t3, t4, o3);
  ins5(t0, t1, t2, t3, t4, o4);

  // Only now take the 5 square roots.
  const float sum5 = sqrtf(fmaxf(t0, 1e-12f)) + sqrtf(fmaxf(t1, 1e-12f)) +
                     sqrtf(fmaxf(t2, 1e-12f)) + sqrtf(fmaxf(t3, 1e-12f)) +
                     sqrtf(fmaxf(t4, 1e-12f));

  // Lane-half 1 duplicates half 0 after the merge; count only half 0.
  float cd = lh ? 0.0f : wvalid * sum5;
  float cw = lh ? 0.0f : wvalid;
#pragma unroll
  for (int off = 16; off > 0; off >>= 1) {
    cd += __shfl_xor(cd, off, 32);
    cw += __shfl_xor(cw, off, 32);
  }

  __shared__ float sdd[WAVES_PB];
  __shared__ float sdw[WAVES_PB];
  if (lane == 0) { sdd[wv] = cd; sdw[wv] = cw; }
  __syncthreads();
  if (threadIdx.x == 0) {
    float td = 0.0f, tw = 0.0f;
#pragma unroll
    for (int i = 0; i < WAVES_PB; ++i) { td += sdd[i]; tw += sdw[i]; }
    partials[2 * blockIdx.x + 0] = td;
    partials[2 * blockIdx.x + 1] = tw;
  }
}

__global__ __launch_bounds__(NBLOCKS)
void knn_final(const float* __restrict__ partials, float* __restrict__ out) {
  __shared__ float sd[NBLOCKS];
  __shared__ float sw[NBLOCKS];
  const int t = threadIdx.x;
  sd[t] = partials[2 * t + 0];
  sw[t] = partials[2 * t + 1];
  __syncthreads();
  for (int s = NBLOCKS / 2; s > 0; s >>= 1) {
    if (t < s) { sd[t] += sd[t + s]; sw[t] += sw[t + s]; }
    __syncthreads();
  }
  if (t == 0) out[0] = sd[0] / (sw[0] * (float)KNN_K);
}

extern "C" void kernel_launch(void* const* d_in, const int* in_sizes, int n_in,
                              void* d_out, int out_size, void* d_ws,
                              size_t ws_size, hipStream_t stream) {
  const float* src = (const float*)d_in[0];   // source_pc, 2*8192*3 f32
  const float* tgt = (const float*)d_in[1];   // target_pc, 2*8192*3 f32
  float* out = (float*)d_out;                 // scalar f32

  // Workspace layout: [0, NPTS) = t2eff, [NPTS, NPTS + 2*NBLOCKS) = partials.
  float* t2eff    = (float*)d_ws;
  float* partials = t2eff + NPTS;

  knn_prep_t2<<<NPTS / 256, 256, 0, stream>>>(tgt, t2eff);
  knn_main<<<NBLOCKS, WAVES_PB * 32, 0, stream>>>(src, tgt, t2eff, partials);
  knn_final<<<1, NBLOCKS, 0, stream>>>(partials, out);
}